// BahdanauAttention_73057393705388
// MI455X (gfx1250) — compile-verified
//
#include <hip/hip_runtime.h>
#include <hip/hip_bf16.h>

typedef __attribute__((ext_vector_type(2))) float v2f;
typedef __attribute__((ext_vector_type(8))) float v8f;

#define BATCH 16
#define TQ    128
#define SV    256
#define DM    256   // d_model == units == 256
#define NEG_INF (-1e9f)
#define LOG2E 1.4426950408889634f

// ---- fast tanh: native v_tanh_f32 on gfx1250, exp2/rcp fallback ----
__device__ __forceinline__ float fast_tanh(float x) {
#if __has_builtin(__builtin_amdgcn_tanhf)
  return __builtin_amdgcn_tanhf(x);
#else
  // tanh(x) = 1 - 2/(exp(2x)+1); exp via v_exp_f32, div via v_rcp_f32.
  float e = __builtin_amdgcn_exp2f(x * (2.0f * LOG2E));
  return 1.0f - 2.0f * __builtin_amdgcn_rcpf(e + 1.0f);
#endif
}

// =====================================================================
// Kernel 1: Y[M x 256] = X[M x 256] * W[256 x 256], f32 WMMA 16x16x4.
// grid = (M/16, 2), block = 256 (8 waves). Wave w owns N-tile
// blockIdx.y*8 + w. A tile (16x256) staged in LDS with stride 260
// (bank-conflict-free fragment reads).
// =====================================================================
__global__ __launch_bounds__(256) void proj_wmma_kernel(
    const float* __restrict__ X, const float* __restrict__ W,
    float* __restrict__ Y) {
  __shared__ float x_lds[16 * 260];

  const int tid   = threadIdx.x;
  const int lane  = tid & 31;
  const int wave  = tid >> 5;
  const int mbase = blockIdx.x * 16;
  const int nbase = (blockIdx.y * 8 + wave) * 16;

  // cooperative load of the 16x256 A tile (coalesced)
  for (int i = tid; i < 16 * 256; i += 256) {
    int r = i >> 8, c = i & 255;
    x_lds[r * 260 + c] = X[(size_t)(mbase + r) * DM + c];
  }
  __syncthreads();

  const int hl  = lane >> 4;   // half-wave select (K offset 0 / 2)
  const int l15 = lane & 15;

  v8f acc = {};
#pragma unroll 4
  for (int ks = 0; ks < 64; ++ks) {
    const int k0 = ks * 4 + 2 * hl;
    v2f a, b;
    a[0] = x_lds[l15 * 260 + k0];
    a[1] = x_lds[l15 * 260 + k0 + 1];
    b[0] = W[(size_t)(k0)     * DM + nbase + l15];
    b[1] = W[(size_t)(k0 + 1) * DM + nbase + l15];
    acc = __builtin_amdgcn_wmma_f32_16x16x4_f32(false, a, false, b,
                                                (short)0, acc, false, false);
  }

#pragma unroll
  for (int v = 0; v < 8; ++v)
    Y[(size_t)(mbase + v + 8 * hl) * DM + nbase + l15] = acc[v];
}

// =====================================================================
// Kernel 2: per (batch, 16-row t-tile):
//   scores -> mask -> softmax -> attn_out, then context = attn * value
//   via f32 WMMA from the LDS score tile.
// grid = (TQ/16, BATCH), block = 256.
// =====================================================================
__global__ __launch_bounds__(256) void bahdanau_attn_kernel(
    const float* __restrict__ Qp,   // (16,128,256) projected query
    const float* __restrict__ Kp,   // (16,256,256) projected value (keys)
    const float* __restrict__ V,    // (16,256,256) raw value
    const unsigned char* __restrict__ mask,  // (16,256) bool
    const float* __restrict__ scale,         // (256,)
    float* __restrict__ ctx_out,    // (16,128,256)
    float* __restrict__ attn_out) { // (16,128,256)
  __shared__ float q_lds[16 * 260];     // padded query tile
  __shared__ float s_lds[16 * 257];     // padded score tile
  __shared__ float scale_lds[256];
  __shared__ float red[16 * 17];
  __shared__ float rowstat[16];

  const int tid = threadIdx.x;
  const int b   = blockIdx.y;
  const int t0  = blockIdx.x * 16;

  for (int i = tid; i < 16 * 256; i += 256) {
    int r = i >> 8, c = i & 255;
    q_lds[r * 260 + c] = Qp[((size_t)b * TQ + t0 + r) * DM + c];
  }
  scale_lds[tid] = scale[tid];
  __syncthreads();

  // ---- additive scores: thread (t, sg) covers s in [sg*16, sg*16+16) ----
  const int t  = tid & 15;
  const int sg = tid >> 4;
  {
    const float4* qrow = (const float4*)(q_lds + t * 260);
    const float4* srow = (const float4*)scale_lds;
    for (int j = 0; j < 16; ++j) {
      const int s = sg * 16 + j;
      const float4* krow = (const float4*)(Kp + ((size_t)b * SV + s) * DM);
      float acc = 0.f;
#pragma unroll 8
      for (int u4 = 0; u4 < 64; ++u4) {
        float4 kv = krow[u4];
        float4 qv = qrow[u4];
        float4 sv = srow[u4];
        acc += sv.x * fast_tanh(qv.x + kv.x);
        acc += sv.y * fast_tanh(qv.y + kv.y);
        acc += sv.z * fast_tanh(qv.z + kv.z);
        acc += sv.w * fast_tanh(qv.w + kv.w);
      }
      s_lds[t * 257 + s] = mask[b * SV + s] ? acc : NEG_INF;
    }
  }
  __syncthreads();

  // ---- masked softmax over s (two-level reduction) ----
  {
    float m = -3.402823466e38f;
    for (int i = 0; i < 16; ++i) m = fmaxf(m, s_lds[t * 257 + sg * 16 + i]);
    red[t * 17 + sg] = m;
  }
  __syncthreads();
  if (tid < 16) {
    float m = red[tid * 17];
    for (int i = 1; i < 16; ++i) m = fmaxf(m, red[tid * 17 + i]);
    rowstat[tid] = m;
  }
  __syncthreads();
  {
    const float m = rowstat[t];
    float psum = 0.f;
    for (int i = 0; i < 16; ++i) {
      float e = __builtin_amdgcn_exp2f((s_lds[t * 257 + sg * 16 + i] - m) * LOG2E);
      s_lds[t * 257 + sg * 16 + i] = e;
      psum += e;
    }
    red[t * 17 + sg] = psum;
  }
  __syncthreads();
  if (tid < 16) {
    float ssum = 0.f;
    for (int i = 0; i < 16; ++i) ssum += red[tid * 17 + i];
    rowstat[tid] = 1.0f / ssum;
  }
  __syncthreads();
  {
    const float inv = rowstat[t];
    for (int i = 0; i < 16; ++i) {
      const int s = sg * 16 + i;
      float p = s_lds[t * 257 + s] * inv;
      s_lds[t * 257 + s] = p;
      attn_out[((size_t)b * TQ + t0 + t) * SV + s] = p;
    }
  }
  __syncthreads();

  // ---- context = attn(16x256) * value(256x256), f32 WMMA ----
  const int lane = tid & 31;
  const int wave = tid >> 5;
  const int hl   = lane >> 4;
  const int l15  = lane & 15;
  const float* Vb = V + (size_t)b * SV * DM;

  for (int nt = wave; nt < 16; nt += 8) {
    const int nbase = nt * 16;
    v8f acc = {};
#pragma unroll 4
    for (int ks = 0; ks < 64; ++ks) {
      const int k0 = ks * 4 + 2 * hl;
      v2f a, bb;
      a[0]  = s_lds[l15 * 257 + k0];
      a[1]  = s_lds[l15 * 257 + k0 + 1];
      bb[0] = Vb[(size_t)(k0)     * DM + nbase + l15];
      bb[1] = Vb[(size_t)(k0 + 1) * DM + nbase + l15];
      acc = __builtin_amdgcn_wmma_f32_16x16x4_f32(false, a, false, bb,
                                                  (short)0, acc, false, false);
    }
#pragma unroll
    for (int v = 0; v < 8; ++v)
      ctx_out[((size_t)b * TQ + t0 + v + 8 * hl) * DM + nbase + l15] = acc[v];
  }
}

extern "C" void kernel_launch(void* const* d_in, const int* in_sizes, int n_in,
                              void* d_out, int out_size, void* d_ws, size_t ws_size,
                              hipStream_t stream) {
  const float*         query = (const float*)d_in[0];   // (16,128,256)
  const float*         value = (const float*)d_in[1];   // (16,256,256)
  const unsigned char* mask  = (const unsigned char*)d_in[2]; // (16,256) bool
  const float*         W1    = (const float*)d_in[3];   // (256,256)
  const float*         W2    = (const float*)d_in[4];   // (256,256)
  const float*         scale = (const float*)d_in[5];   // (256,)

  float* ctx  = (float*)d_out;                 // context first
  float* attn = ctx + (size_t)BATCH * TQ * DM; // then attention weights

  float* qp = (float*)d_ws;                         // (16,128,256)
  float* kp = qp + (size_t)BATCH * TQ * DM;         // (16,256,256)

  // q = query @ W1  (2048 rows)
  proj_wmma_kernel<<<dim3(TQ * BATCH / 16, 2), 256, 0, stream>>>(query, W1, qp);
  // k = value @ W2  (4096 rows)
  proj_wmma_kernel<<<dim3(SV * BATCH / 16, 2), 256, 0, stream>>>(value, W2, kp);
  // scores -> softmax -> context
  bahdanau_attn_kernel<<<dim3(TQ / 16, BATCH), 256, 0, stream>>>(
      qp, kp, value, mask, scale, ctx, attn);
}